// ExerciseNet_51857435132234
// MI455X (gfx1250) — compile-verified
//
#include <hip/hip_runtime.h>
#include <hip/hip_bf16.h>
#include <math.h>

// Problem constants (from reference)
#define WCNT 50000
#define EMB  25
#define H    25
#define G3   75      // 3*H
#define TT   200
#define BB   4096
#define BT   16      // batch tile (M of WMMA)
#define KP   28      // K padded to multiple of 4 (25 -> 28)
#define NP   80      // 3H padded to multiple of 16 (75 -> 80)
#define NT   5       // N tiles of 16
#define KT   7       // K steps of 4
#define NWV  10      // waves per block: {x-proj, h-proj} x 5 N-tiles
#define THREADS (NWV * 32)
#define NW2  2       // ceil(BT*H / THREADS) gate elements per thread

typedef __attribute__((ext_vector_type(2))) float v2f;
typedef __attribute__((ext_vector_type(8))) float v8f;

// Fast gate nonlinearities: one v_exp_f32 + one v_rcp_f32 each (TRANS pipe),
// instead of the IEEE div_scale/div_fmas sequence.
__device__ __forceinline__ float sigmoidf_(float x) {
    return __builtin_amdgcn_rcpf(1.0f + __expf(-x));
}
// tanh(x) = 1 - 2/(1 + e^{2x})
__device__ __forceinline__ float tanhf_(float x) {
    return 1.0f - 2.0f * __builtin_amdgcn_rcpf(1.0f + __expf(2.0f * x));
}

// 10 wave32 per block. blockIdx.x = batch tile, blockIdx.y = direction.
// Wave w computes one 16-column tile of ONE of the two per-step GEMMs
// (x-projection or h-projection) on the fp32 WMMA pipe; gate math is
// distributed over all 320 threads. GRU state lives in LDS for the whole
// 200-step scan; weight fragments live in registers (14 VGPRs per wave).
__global__ __launch_bounds__(THREADS) void bigru_scan_kernel(
    const int*   __restrict__ inp,     // [T, B]
    const float* __restrict__ h0,      // [2, B, H]
    const float* __restrict__ emb,     // [WCNT, EMB] (row 0 is zero = padding)
    const float* __restrict__ wih_f, const float* __restrict__ whh_f,
    const float* __restrict__ bih_f, const float* __restrict__ bhh_f,
    const float* __restrict__ wih_b, const float* __restrict__ whh_b,
    const float* __restrict__ bih_b, const float* __restrict__ bhh_b,
    float* __restrict__ out)           // [B,2H] y  then [2,B,H] h
{
    const int tid   = threadIdx.x;
    const int wid   = tid >> 5;
    const int lane  = tid & 31;
    const int dir   = blockIdx.y;
    const int bbase = blockIdx.x * BT;

    const int mat = (wid >= NT) ? 1 : 0;   // 0 = x-projection, 1 = h-projection
    const int nt  = mat ? (wid - NT) : wid;

    const float* wih = dir ? wih_b : wih_f;
    const float* whh = dir ? whh_b : whh_f;
    const float* bih = dir ? bih_b : bih_f;
    const float* bhh = dir ? bhh_b : bhh_f;

    __shared__ float Bx[KP * NP];    // w_ih as B-matrix: Bx[k][n] = wih[n*EMB+k]
    __shared__ float Bh[KP * NP];    // w_hh as B-matrix
    __shared__ float hs[BT * KP];    // GRU state (K-padded with zeros)
    __shared__ float preX[BT * NP];  // x@wih^T + bih
    __shared__ float preH[BT * NP];  // h@whh^T + bhh

    // Zero-init padded LDS (padding must stay 0 for WMMA correctness)
    for (int e = tid; e < KP * NP; e += THREADS) { Bx[e] = 0.0f; Bh[e] = 0.0f; }
    for (int e = tid; e < BT * KP; e += THREADS) { hs[e] = 0.0f; }
    __syncthreads();

    // Stage weights transposed into LDS: B[k][n] layout, n = gate index
    for (int e = tid; e < G3 * EMB; e += THREADS) { int g = e / EMB, i = e % EMB; Bx[i * NP + g] = wih[e]; }
    for (int e = tid; e < G3 * H;   e += THREADS) { int g = e / H,   j = e % H;   Bh[j * NP + g] = whh[e]; }

    // Initial hidden state
    for (int e = tid; e < BT * H; e += THREADS) {
        int m = e / H, j = e % H;
        hs[m * KP + j] = h0[dir * BB * H + (bbase + m) * H + j];
    }
    __syncthreads();

    // Per-lane constants for fragment addressing (wave32 WMMA layouts)
    const int col   = lane & 15;         // N column (and A-frag row M)
    const int khalf = (lane >> 4) * 2;   // upper half-wave handles K+2,K+3
    const int mhi   = 8 * (lane >> 4);   // C/D row offset for upper half-wave
    const int g     = nt * 16 + col;     // this wave's output column

    // This wave's weight B-fragments: loop-invariant, 14 VGPRs
    const float* Bm = mat ? Bh : Bx;
    v2f bw[KT];
#pragma unroll
    for (int kk = 0; kk < KT; ++kk) {
        const int kb = kk * 4 + khalf;
        bw[kk].x = Bm[kb * NP + g];
        bw[kk].y = Bm[(kb + 1) * NP + g];
    }
    // This wave's bias column (folded into the scatter add)
    const float bs = (g < G3) ? (mat ? bhh[g] : bih[g]) : 0.0f;

    float* plane = mat ? preH : preX;

    // Running max over time in registers (thread owns fixed (m,j) elements)
    float ym[NW2];
#pragma unroll
    for (int i = 0; i < NW2; ++i) ym[i] = -INFINITY;

    const v8f zero = {0.0f, 0.0f, 0.0f, 0.0f, 0.0f, 0.0f, 0.0f, 0.0f};

    for (int t = 0; t < TT; ++t) {
        const int tt = dir ? (TT - 1 - t) : t;

        // A-fragments: x-waves gather the embedding row for batch row m=col
        // straight from global (emb is L2-resident); h-waves read the LDS
        // state. Branch condition is wave-uniform -> EXEC stays all-ones.
        v2f af[KT];
        if (mat == 0) {
            const int myidx = inp[tt * BB + bbase + col];
            const float* xrow = emb + (size_t)myidx * EMB;
#pragma unroll
            for (int kk = 0; kk < KT - 1; ++kk) {        // k <= 23: always in range
                const int k0 = kk * 4 + khalf;
                af[kk].x = xrow[k0];
                af[kk].y = xrow[k0 + 1];
            }
            // kk=6 touches K = 24..27; only (khalf==0, .x) is real data.
            const float v24 = xrow[24];                   // unconditional, in-bounds
            af[KT - 1].x = (khalf == 0) ? v24 : 0.0f;
            af[KT - 1].y = 0.0f;
        } else {
#pragma unroll
            for (int kk = 0; kk < KT; ++kk) {             // padded cols stay zero
                const int k0 = kk * 4 + khalf;
                af[kk].x = hs[col * KP + k0];
                af[kk].y = hs[col * KP + k0 + 1];
            }
        }

        // One 16x16 tile, K-reduction of 7 fp32 WMMAs (C starts at inline 0)
        v8f acc = zero;
#pragma unroll
        for (int kk = 0; kk < KT; ++kk)
            acc = __builtin_amdgcn_wmma_f32_16x16x4_f32(false, af[kk], false, bw[kk],
                                                        (short)0, acc, false, false);

        // Scatter this wave's tile (+bias) to its pre-activation plane
#pragma unroll
        for (int v = 0; v < 8; ++v)
            plane[(v + mhi) * NP + g] = acc[v] + bs;
        __syncthreads();

        // Gate math + state update + running max, spread over all 320 threads
#pragma unroll
        for (int i = 0; i < NW2; ++i) {
            const int e = tid + i * THREADS;
            if (e < BT * H) {
                const int m = e / H, j = e % H;
                const float r  = sigmoidf_(preX[m * NP + j]         + preH[m * NP + j]);
                const float z  = sigmoidf_(preX[m * NP + H + j]     + preH[m * NP + H + j]);
                const float hn = preH[m * NP + 2 * H + j];
                const float nn = tanhf_(preX[m * NP + 2 * H + j] + r * hn);
                const float hv = (1.0f - z) * nn + z * hs[m * KP + j];
                hs[m * KP + j] = hv;
                ym[i] = fmaxf(ym[i], hv);
            }
        }
        __syncthreads();
    }

    // Outputs: y [B, 2H] (fwd cols 0..24, bwd cols 25..49), then h [2, B, H]
#pragma unroll
    for (int i = 0; i < NW2; ++i) {
        const int e = tid + i * THREADS;
        if (e < BT * H) {
            const int m = e / H, j = e % H;
            const int b = bbase + m;
            out[b * (2 * H) + dir * H + j] = ym[i];
            out[2 * H * BB + dir * BB * H + b * H + j] = hs[m * KP + j];
        }
    }
}

extern "C" void kernel_launch(void* const* d_in, const int* in_sizes, int n_in,
                              void* d_out, int out_size, void* d_ws, size_t ws_size,
                              hipStream_t stream) {
    (void)in_sizes; (void)n_in; (void)d_ws; (void)ws_size; (void)out_size;
    const int*   inp  = (const int*)  d_in[0];
    const float* hid  = (const float*)d_in[1];
    const float* emb  = (const float*)d_in[2];
    const float* wihf = (const float*)d_in[3];
    const float* whhf = (const float*)d_in[4];
    const float* bihf = (const float*)d_in[5];
    const float* bhhf = (const float*)d_in[6];
    const float* wihb = (const float*)d_in[7];
    const float* whhb = (const float*)d_in[8];
    const float* bihb = (const float*)d_in[9];
    const float* bhhb = (const float*)d_in[10];

    dim3 grid(BB / BT, 2);   // 256 batch tiles x 2 directions
    bigru_scan_kernel<<<grid, THREADS, 0, stream>>>(
        inp, hid, emb,
        wihf, whhf, bihf, bhhf,
        wihb, whhb, bihb, bhhb,
        (float*)d_out);
}